// UncertaintyAutogradLoss_56934086476281
// MI455X (gfx1250) — compile-verified
//
#include <hip/hip_runtime.h>
#include <math.h>

#define BLOCK 256

typedef __attribute__((ext_vector_type(4))) float v4f;
typedef __attribute__((ext_vector_type(2))) float v2f;

// Closed-form Gaussian-NLL value / gradient / Hessian wrt (mu, b=pre-softplus std).
// Uses one v_exp, two v_log, two v_rcp per element; no libm calls, no IEEE divs.
__device__ __forceinline__ void nll_all(float mu, float b, float y,
                                        float& loss, float& g0, float& g1,
                                        float& h00, float& h01, float& h11) {
    float ab  = fabsf(b);
    float e   = __expf(-ab);                   // exp(-|b|)  (v_exp_f32)
    float t   = 1.0f + e;
    float sp  = fmaxf(b, 0.0f) + __logf(t);    // stable softplus; log1p->log ok: corr term tiny vs sp
    float rt  = __builtin_amdgcn_rcpf(t);      // 1/(1+e)    (v_rcp_f32)
    float sig = (b >= 0.0f) ? rt : e * rt;     // sigmoid(b)
    float inv_s = __builtin_amdgcn_rcpf(sp);   // 1/softplus (v_rcp_f32)

    float z  = (y - mu) * inv_s;
    float z2 = z * z;
    loss = 0.5f * z2 + __logf(sp) + 0.91893853320467274178f; // + 0.5*log(2*pi)
    g0   = -z * inv_s;
    float dfds = (1.0f - z2) * inv_s;          // df/ds
    g1   = dfds * sig;
    float is2 = inv_s * inv_s;
    h00  = is2;
    h01  = 2.0f * z * is2 * sig;
    h11  = (3.0f * z2 - 1.0f) * is2 * sig * sig + dfds * sig * (1.0f - sig);
}

__global__ void __launch_bounds__(BLOCK)
nll_fwd_grad_hess_kernel(const float* __restrict__ pred,   // [N,2]
                         const float* __restrict__ ytrue,  // [N]
                         float* __restrict__ out,          // [N] | [N,2] | [N,2,2]
                         int n, int iters) {
    // Double-buffered LDS staging: 16B pred + 8B ytrue per thread per buffer.
    __shared__ v4f sPred[2][BLOCK];
    __shared__ v2f sTrue[2][BLOCK];

    const int       tid     = threadIdx.x;
    const long long gthread = (long long)blockIdx.x * BLOCK + tid;
    const long long tstride = (long long)gridDim.x * BLOCK;

    // Flat LDS byte offsets (low 32 bits of the shared-aperture flat address).
    unsigned lp[2], lt[2];
    lp[0] = (unsigned)(size_t)&sPred[0][tid];
    lp[1] = (unsigned)(size_t)&sPred[1][tid];
    lt[0] = (unsigned)(size_t)&sTrue[0][tid];
    lt[1] = (unsigned)(size_t)&sTrue[1][tid];

    float* lossO = out;
    float* gradO = out + (size_t)n;
    float* hessO = out + (size_t)3 * n;

    // ---- prologue: async-copy tile 0 into buffer 0 ----
    {
        long long e0 = gthread * 2;
        if (e0 + 1 < n) {
            unsigned offp = (unsigned)(e0 * 8); // 2 floats/elem
            unsigned offt = (unsigned)(e0 * 4);
            asm volatile("global_load_async_to_lds_b128 %0, %1, %2\n\t"
                         "global_load_async_to_lds_b64  %3, %4, %5"
                         :: "v"(lp[0]), "v"(offp), "s"(pred),
                            "v"(lt[0]), "v"(offt), "s"(ytrue)
                         : "memory");
        }
    }

    for (int k = 0; k < iters; ++k) {
        const int       cur  = k & 1;
        const long long e0   = (gthread + (long long)k * tstride) * 2;
        const bool      full = (e0 + 1) < (long long)n;

        // ---- prefetch tile k+1 into the other buffer while tile k is consumed ----
        if (k + 1 < iters) {
            long long en = (gthread + (long long)(k + 1) * tstride) * 2;
            if (en + 1 < n) {
                const int nb = cur ^ 1;
                unsigned offp = (unsigned)(en * 8);
                unsigned offt = (unsigned)(en * 4);
                // s_wait_dscnt 0: prior-iteration LDS reads must land before we
                // overwrite that buffer via the async path.
                asm volatile("s_wait_dscnt 0\n\t"
                             "global_load_async_to_lds_b128 %0, %1, %2\n\t"
                             "global_load_async_to_lds_b64  %3, %4, %5"
                             :: "v"(lp[nb]), "v"(offp), "s"(pred),
                                "v"(lt[nb]), "v"(offt), "s"(ytrue)
                             : "memory");
            }
            // 2 copies of the next tile may remain in flight; current tile done.
            asm volatile("s_wait_asynccnt 2" ::: "memory");
        } else {
            asm volatile("s_wait_asynccnt 0" ::: "memory");
        }

        if (full) {
            v4f p  = sPred[cur][tid];   // (mu0,b0,mu1,b1)
            v2f yv = sTrue[cur][tid];

            float l0, a0, a1, c00, c01, c11;
            float l1, b0, b1, d00, d01, d11;
            nll_all(p.x, p.y, yv.x, l0, a0, a1, c00, c01, c11);
            nll_all(p.z, p.w, yv.y, l1, b0, b1, d00, d01, d11);

            v2f lv  = {l0, l1};
            v4f gv  = {a0, a1, b0, b1};
            v4f h0v = {c00, c01, c01, c11};
            v4f h1v = {d00, d01, d01, d11};
            // Streaming outputs (336MB > 192MB L2): non-temporal B64/B128 stores.
            __builtin_nontemporal_store(lv,  (v2f*)(lossO + e0));
            __builtin_nontemporal_store(gv,  (v4f*)(gradO + e0 * 2));
            __builtin_nontemporal_store(h0v, (v4f*)(hessO + e0 * 4));
            __builtin_nontemporal_store(h1v, (v4f*)(hessO + e0 * 4 + 4));
        } else if (e0 < (long long)n) {
            // scalar tail (at most one element grid-wide)
            float mu = pred[e0 * 2], bb = pred[e0 * 2 + 1], y = ytrue[e0];
            float l, g0s, g1s, h00, h01, h11;
            nll_all(mu, bb, y, l, g0s, g1s, h00, h01, h11);
            lossO[e0]         = l;
            gradO[e0 * 2]     = g0s;
            gradO[e0 * 2 + 1] = g1s;
            hessO[e0 * 4]     = h00;
            hessO[e0 * 4 + 1] = h01;
            hessO[e0 * 4 + 2] = h01;
            hessO[e0 * 4 + 3] = h11;
        }
    }
}

extern "C" void kernel_launch(void* const* d_in, const int* in_sizes, int n_in,
                              void* d_out, int out_size, void* d_ws, size_t ws_size,
                              hipStream_t stream) {
    const float* pred  = (const float*)d_in[0]; // [N,2]
    const float* ytrue = (const float*)d_in[1]; // [N]
    float*       out   = (float*)d_out;         // loss[N] | grad[N,2] | hess[N,2,2]

    const long long n      = (long long)in_sizes[1];
    const long long npairs = (n + 1) / 2;

    long long blocksNeeded = (npairs + BLOCK - 1) / BLOCK;
    int blocks = (int)(blocksNeeded < 4096 ? blocksNeeded : 4096);
    if (blocks < 1) blocks = 1;
    const long long tstride = (long long)blocks * BLOCK;
    int iters = (int)((npairs + tstride - 1) / tstride);
    if (iters < 1) iters = 1;

    nll_fwd_grad_hess_kernel<<<blocks, BLOCK, 0, stream>>>(pred, ytrue, out, (int)n, iters);
}